// GroupedRecurrentMultiHeadAttention_30597347017215
// MI455X (gfx1250) — compile-verified
//
#include <hip/hip_runtime.h>
#include <math.h>

typedef float v2f __attribute__((ext_vector_type(2)));
typedef float v8f __attribute__((ext_vector_type(8)));

// ---------------------------------------------------------------- constants
#define BB 2
#define SS 2048
#define DM 1024
#define KV 512
#define QH 16
#define DH 64
#define ROWS (BB*SS)          // 4096
#define QROWS (BB*SS*2)       // 8192 rows of sigma_q viewed (.,512)

// workspace layout (in floats)
#define OFF_Q    ((size_t)0)
#define OFF_K    ((size_t)4194304)
#define OFF_V    ((size_t)6291456)
#define OFF_ADOT ((size_t)8388608)
#define OFF_U    ((size_t)12582912)
#define OFF_DENQ ((size_t)14680064)
#define OFF_DENK ((size_t)14688256)

// output layout (in floats)
#define OUT_MEM  ((size_t)4194304)
#define OUT_NORM ((size_t)4456448)

static __device__ __forceinline__ float eluf(float x) {
    return x > 0.0f ? x : expm1f(x);
}

static __device__ __forceinline__ v8f wmma4(v2f a, v2f b, v8f c) {
    // D = A(16x4,f32) * B(4x16,f32) + C(16x16,f32)
    return __builtin_amdgcn_wmma_f32_16x16x4_f32(
        false, a, false, b, (short)0, c, false, false);
}

// ---------------------------------------------------------------- generic WMMA GEMM
// One wave computes one 16x64 strip (4 C-tiles) of C = opA(A) * B (+ epilogue).
// EPI 0: C = acc
// EPI 1: C = aux - acc/den[row]            (u = v - retrieved)
// EPI 2: C = aux + acc                     (new_memory = memory + skT@u)
// EPI 3: C = aux*(1-w) + (acc/den[row])*w  (gated fusion, w = sigmoid(mw[0]))
template<bool ELU_A, bool TRANS_A, int EPI>
__global__ __launch_bounds__(256) void gemm_wmma(
    const float* __restrict__ A, const float* __restrict__ Bm, float* __restrict__ C,
    int M, int N, int K, int lda, int ldb, int ldc,
    const float* __restrict__ den, const float* __restrict__ aux,
    const float* __restrict__ mw)
{
    const int wave = threadIdx.x >> 5;
    const int lane = threadIdx.x & 31;
    const int hi   = lane >> 4;        // 0: lanes 0-15, 1: lanes 16-31
    const int l16  = lane & 15;
    const int strip = blockIdx.x * 8 + wave;   // 64-column strip index
    const int tm = blockIdx.y;
    if (strip >= (N >> 6)) return;     // wave-uniform: EXEC stays all-ones
    const int mbase = tm * 16;
    const int row = mbase + l16;
    const int nb0 = strip * 64;

    v8f acc[4];
#pragma unroll
    for (int n = 0; n < 4; ++n) acc[n] = (v8f){0.f,0.f,0.f,0.f,0.f,0.f,0.f,0.f};

#pragma unroll 4
    for (int k0 = 0; k0 < K; k0 += 4) {
        const int ka = k0 + 2*hi;      // K index held by this half (vgpr0); vgpr1 = ka+1
        v2f a;
        if (TRANS_A) {
            a[0] = A[(size_t)ka     * lda + row];
            a[1] = A[(size_t)(ka+1) * lda + row];
        } else {
            a[0] = A[(size_t)row * lda + ka];
            a[1] = A[(size_t)row * lda + ka + 1];
        }
        if (ELU_A) { a[0] = eluf(a[0]); a[1] = eluf(a[1]); }
#pragma unroll
        for (int n = 0; n < 4; ++n) {
            const int col = nb0 + n*16 + l16;
            v2f b;
            b[0] = Bm[(size_t)ka     * ldb + col];
            b[1] = Bm[(size_t)(ka+1) * ldb + col];
            acc[n] = wmma4(a, b, acc[n]);
        }
    }

    float w = 0.0f;
    if (EPI == 3) w = 1.0f / (1.0f + expf(-mw[0]));
#pragma unroll
    for (int n = 0; n < 4; ++n) {
#pragma unroll
        for (int r = 0; r < 8; ++r) {
            const int orow = mbase + r + 8*hi;
            const int col  = nb0 + n*16 + l16;
            const size_t idx = (size_t)orow * ldc + col;
            const float val = acc[n][r];
            if      (EPI == 0) C[idx] = val;
            else if (EPI == 1) C[idx] = aux[idx] - val / den[orow];
            else if (EPI == 2) C[idx] = aux[idx] + val;
            else               C[idx] = aux[idx] * (1.0f - w) + (val / den[orow]) * w;
        }
    }
}

// ---------------------------------------------------------------- flash attention
// One wave per (batch, flat-head, 16-row q tile). Online softmax over S.
__global__ __launch_bounds__(32) void attn_kernel(
    const float* __restrict__ q, const float* __restrict__ k,
    const float* __restrict__ v, float* __restrict__ adot)
{
    const int lane = threadIdx.x & 31;
    const int hi   = lane >> 4;
    const int l16  = lane & 15;
    const int qt = blockIdx.x;     // 0..127
    const int h  = blockIdx.y;     // 0..15
    const int b  = blockIdx.z;     // 0..1
    const float scale = 0.125f;    // 1/sqrt(64)
    const int bS = b * SS;
    const float* qrow = q + (size_t)(bS + qt*16 + l16) * DM + h * DH;
    const int hk = (h >> 1) * DH;
    const float* kb = k + (size_t)bS * KV + hk;
    const float* vb = v + (size_t)bS * KV + hk;

    // preload q A-fragments for all 16 K-steps (D_HEAD=64)
    v2f qa[16];
#pragma unroll
    for (int kk = 0; kk < 16; ++kk) {
        const int d = kk*4 + 2*hi;
        qa[kk][0] = qrow[d];
        qa[kk][1] = qrow[d+1];
    }

    float m_r[8], l_r[8];
    v8f acc[4];
#pragma unroll
    for (int r = 0; r < 8; ++r) { m_r[r] = -INFINITY; l_r[r] = 0.0f; }
#pragma unroll
    for (int n = 0; n < 4; ++n) acc[n] = (v8f){0.f,0.f,0.f,0.f,0.f,0.f,0.f,0.f};

    __shared__ float pbuf[16 * 17];   // 17-stride pad to dodge bank conflicts

    for (int t0 = 0; t0 < SS; t0 += 16) {
        // scores tile: S = (q @ k^T) * scale, 16 WMMA K-steps
        v8f s = {0.f,0.f,0.f,0.f,0.f,0.f,0.f,0.f};
#pragma unroll
        for (int kk = 0; kk < 16; ++kk) {
            const int d = kk*4 + 2*hi;
            const float* kp = kb + (size_t)(t0 + l16) * KV + d;  // B[d][col]=k[t][d]
            v2f bf; bf[0] = kp[0]; bf[1] = kp[1];
            s = wmma4(qa[kk], bf, s);
        }
        // online softmax (rows live across the 16 lanes of each half-wave)
        float alpha[8], p[8];
#pragma unroll
        for (int r = 0; r < 8; ++r) {
            float x = s[r] * scale;
            float mx = x;
            mx = fmaxf(mx, __shfl_xor(mx, 1, 32));
            mx = fmaxf(mx, __shfl_xor(mx, 2, 32));
            mx = fmaxf(mx, __shfl_xor(mx, 4, 32));
            mx = fmaxf(mx, __shfl_xor(mx, 8, 32));
            const float nm = fmaxf(m_r[r], mx);
            const float al = __expf(m_r[r] - nm);
            const float pe = __expf(x - nm);
            float rs = pe;
            rs += __shfl_xor(rs, 1, 32);
            rs += __shfl_xor(rs, 2, 32);
            rs += __shfl_xor(rs, 4, 32);
            rs += __shfl_xor(rs, 8, 32);
            l_r[r] = l_r[r] * al + rs;
            m_r[r] = nm;
            alpha[r] = al;
            p[r] = pe;
        }
#pragma unroll
        for (int n = 0; n < 4; ++n)
#pragma unroll
            for (int r = 0; r < 8; ++r) acc[n][r] *= alpha[r];

        // stage P (C-layout) -> LDS -> reload as A-fragments
        __syncthreads();
#pragma unroll
        for (int r = 0; r < 8; ++r) pbuf[(r + 8*hi)*17 + l16] = p[r];
        __syncthreads();

        // acc += P @ V  (4 K-steps x 4 N-tiles)
#pragma unroll
        for (int kk = 0; kk < 4; ++kk) {
            const int d = kk*4 + 2*hi;
            v2f pa; pa[0] = pbuf[l16*17 + d]; pa[1] = pbuf[l16*17 + d + 1];
#pragma unroll
            for (int n = 0; n < 4; ++n) {
                const float* vp = vb + (size_t)(t0 + d) * KV + n*16 + l16;
                v2f bf; bf[0] = vp[0]; bf[1] = vp[KV];
                acc[n] = wmma4(pa, bf, acc[n]);
            }
        }
    }

#pragma unroll
    for (int n = 0; n < 4; ++n)
#pragma unroll
        for (int r = 0; r < 8; ++r) {
            const int orow = qt*16 + r + 8*hi;
            adot[(size_t)(bS + orow) * DM + h*DH + n*16 + l16] = acc[n][r] / l_r[r];
        }
}

// ---------------------------------------------------------------- denominators
// rows 0..8191: den_q over q-as-(8192,512); rows 8192..12287: den_k over k
__global__ __launch_bounds__(256) void row_dots(
    const float* __restrict__ q, const float* __restrict__ k,
    const float* __restrict__ mnorm, float* __restrict__ denq, float* __restrict__ denk)
{
    const int wave = threadIdx.x >> 5;
    const int lane = threadIdx.x & 31;
    const int row = blockIdx.x * 8 + wave;
    const float* src = (row < QROWS) ? (q + (size_t)row * KV)
                                     : (k + (size_t)(row - QROWS) * KV);
    float s = 0.0f;
    for (int c = lane; c < KV; c += 32) s += eluf(src[c]) * mnorm[c];
    s += __shfl_xor(s, 1, 32);
    s += __shfl_xor(s, 2, 32);
    s += __shfl_xor(s, 4, 32);
    s += __shfl_xor(s, 8, 32);
    s += __shfl_xor(s, 16, 32);
    if (lane == 0) {
        if (row < QROWS) denq[row] = s;
        else             denk[row - QROWS] = s;
    }
}

// new_memory_norm[j] = memory_norm[j] + sum_rows elu(k[r][j])
__global__ __launch_bounds__(256) void colsum_norm(
    const float* __restrict__ k, const float* __restrict__ mnorm,
    float* __restrict__ out_nm)
{
    const int col = blockIdx.x * 256 + threadIdx.x;
    if (col >= KV) return;
    float s = mnorm[col];
    for (int r = 0; r < ROWS; ++r) s += eluf(k[(size_t)r * KV + col]);
    out_nm[col] = s;
}

// ---------------------------------------------------------------- launcher
extern "C" void kernel_launch(void* const* d_in, const int* in_sizes, int n_in,
                              void* d_out, int out_size, void* d_ws, size_t ws_size,
                              hipStream_t stream) {
    const float* x      = (const float*)d_in[0];
    const float* Wq     = (const float*)d_in[1];
    const float* Wk     = (const float*)d_in[2];
    const float* Wv     = (const float*)d_in[3];
    const float* memory = (const float*)d_in[4];
    const float* mnorm  = (const float*)d_in[5];
    const float* mw     = (const float*)d_in[6];
    float* out = (float*)d_out;

    float* W = (float*)d_ws;
    float* qb   = W + OFF_Q;
    float* kb   = W + OFF_K;
    float* vb   = W + OFF_V;
    float* adot = W + OFF_ADOT;
    float* ub   = W + OFF_U;
    float* denq = W + OFF_DENQ;
    float* denk = W + OFF_DENK;

    // 1-3. q/k/v projections (strips of 64 cols: grid.x = N/64/8)
    gemm_wmma<false,false,0><<<dim3(2,256), 256, 0, stream>>>(
        x, Wq, qb, ROWS, DM, DM, DM, DM, DM, nullptr, nullptr, nullptr);
    gemm_wmma<false,false,0><<<dim3(1,256), 256, 0, stream>>>(
        x, Wk, kb, ROWS, KV, DM, DM, KV, KV, nullptr, nullptr, nullptr);
    gemm_wmma<false,false,0><<<dim3(1,256), 256, 0, stream>>>(
        x, Wv, vb, ROWS, KV, DM, DM, KV, KV, nullptr, nullptr, nullptr);

    // 4. grouped attention -> a_dot
    attn_kernel<<<dim3(SS/16, QH, BB), 32, 0, stream>>>(qb, kb, vb, adot);

    // 5. denominators den_q (8192), den_k (4096)
    row_dots<<<dim3((QROWS + ROWS)/8), 256, 0, stream>>>(qb, kb, mnorm, denq, denk);

    // 6. u = v - (elu(k) @ memory) / den_k      [M=4096,N=512,K=512]
    gemm_wmma<true,false,1><<<dim3(1,256), 256, 0, stream>>>(
        kb, memory, ub, ROWS, KV, KV, KV, KV, KV, denk, vb, nullptr);

    // 7. new_memory = memory + elu(k)^T @ u     [M=512,N=512,K=4096]
    gemm_wmma<true,true,2><<<dim3(1,32), 256, 0, stream>>>(
        kb, ub, out + OUT_MEM, KV, KV, ROWS, KV, KV, KV, nullptr, memory, nullptr);

    // 8. new_memory_norm
    colsum_norm<<<dim3(2), 256, 0, stream>>>(kb, mnorm, out + OUT_NORM);

    // 9. out = a_dot*(1-w) + (elu(q)@memory / den_q)*w   [M=8192,N=512,K=512]
    gemm_wmma<true,false,3><<<dim3(1,512), 256, 0, stream>>>(
        qb, memory, out, QROWS, KV, KV, KV, KV, KV, denq, adot, mw);
}